// EncoderLayer_18159121727552
// MI455X (gfx1250) — compile-verified
//
#include <hip/hip_runtime.h>
#include <math.h>

typedef __attribute__((ext_vector_type(2))) float v2f;
typedef __attribute__((ext_vector_type(8))) float v8f;

#define BM 128
#define BN 64
#define KB 16
#define KP (KB + 2)      // padded pitch: 18 floats -> 8B-aligned K-pairs
#define EPSF 1e-6f

// ---------------------------------------------------------------------------
// Generic WMMA fp32 GEMM:  C = scale*(A@B) [+bias] [+residual] [relu]
// flags: 1 = B transposed (B stored [N x K] row-major, i.e. C=A@B^T)
//        2 = relu epilogue
//        4 = block-diagonal B (per-64-col head: B + (n0/64)*64*64, A col base=n0)
// Block 256 = 8 waves over a 128x64 C block; each wave owns 32x32 (4 WMMA
// accumulators). K staged through double-buffered LDS tiles; A is k-minor
// (a_lds[m][k]), B is n-major (b_lds[n][k]) so every fragment is one b64 load.
// ---------------------------------------------------------------------------
__global__ __launch_bounds__(256) void gemm_wmma_f32(
    const float* __restrict__ A, const float* __restrict__ B,
    const float* __restrict__ bias, const float* __restrict__ residual,
    float* __restrict__ C,
    int M, int N, int K, int lda, int ldb, int ldc, float scale, int flags)
{
  __shared__ float a_lds[2][BM][KP];
  __shared__ float b_lds[2][BN][KP];

  const int t  = threadIdx.x;
  const int l  = t & 31;
  const int w  = t >> 5;
  const int wm = w & 3;              // 4 row tiles of 32
  const int wn = w >> 2;             // 2 col tiles of 32
  const int m0 = blockIdx.y * BM;
  const int n0 = blockIdx.x * BN;

  const bool bT   = (flags & 1) != 0;
  const bool relu = (flags & 2) != 0;
  const bool bd   = (flags & 4) != 0;

  const float* Bp = B + (bd ? (size_t)(n0 >> 6) * 64 * 64 : 0);
  const int acol = bd ? n0 : 0;      // block-diagonal: A columns follow C cols
  const int bn0  = bd ? 0 : n0;      // block-diagonal: B is a private head

  // A staging: each thread 8 floats of one row
  const int ar = t >> 1;             // 0..127
  const int ak = (t & 1) * 8;        // 0 or 8
  // B staging (NN): row k = t>>4, 4 cols; (T): row n = t>>2, 4 ks
  const int bkr = t >> 4;            // 0..15
  const int bnv = (t & 15) * 4;      // 0..60
  const int btn = t >> 2;            // 0..63
  const int btk = (t & 3) * 4;       // 0..12

  v8f c00 = {}, c01 = {}, c10 = {}, c11 = {};
  float4 pa0, pa1, pb;

  // ---- global tile fetch into registers ----
  auto fetch = [&](int k0) {
    const float* ap = A + (size_t)(m0 + ar) * lda + acol + k0 + ak;
    pa0 = *(const float4*)(ap);
    pa1 = *(const float4*)(ap + 4);
    if (!bT) pb = *(const float4*)(Bp + (size_t)(k0 + bkr) * ldb + bn0 + bnv);
    else     pb = *(const float4*)(Bp + (size_t)(n0 + btn) * ldb + k0 + btk);
  };
  // ---- registers -> LDS ----
  auto stage = [&](int buf) {
    a_lds[buf][ar][ak + 0] = pa0.x;
    a_lds[buf][ar][ak + 1] = pa0.y;
    a_lds[buf][ar][ak + 2] = pa0.z;
    a_lds[buf][ar][ak + 3] = pa0.w;
    a_lds[buf][ar][ak + 4] = pa1.x;
    a_lds[buf][ar][ak + 5] = pa1.y;
    a_lds[buf][ar][ak + 6] = pa1.z;
    a_lds[buf][ar][ak + 7] = pa1.w;
    if (!bT) {                       // transpose k-major global -> n-major LDS
      b_lds[buf][bnv + 0][bkr] = pb.x;
      b_lds[buf][bnv + 1][bkr] = pb.y;
      b_lds[buf][bnv + 2][bkr] = pb.z;
      b_lds[buf][bnv + 3][bkr] = pb.w;
    } else {                         // key_sm rows are already n-major
      b_lds[buf][btn][btk + 0] = pb.x;
      b_lds[buf][btn][btk + 1] = pb.y;
      b_lds[buf][btn][btk + 2] = pb.z;
      b_lds[buf][btn][btk + 3] = pb.w;
    }
  };

  fetch(0);
  stage(0);
  __syncthreads();

  const int am  = wm * 32 + (l & 15);     // A fragment row (this wave)
  const int bn  = wn * 32 + (l & 15);     // B fragment col (this wave)
  const int half = (l >> 4) * 2;          // lanes 0-15: K+{0,1}; 16-31: K+{2,3}

  int buf = 0;
  for (int k0 = 0; k0 < K; k0 += KB) {
    const bool more = (k0 + KB) < K;
    if (more) fetch(k0 + KB);            // overlap next global tile with math

#pragma unroll
    for (int k4 = 0; k4 < KB; k4 += 4) {
      v2f a0, a1, b0, b1;
      a0.x = a_lds[buf][am][k4 + half];
      a0.y = a_lds[buf][am][k4 + half + 1];
      a1.x = a_lds[buf][am + 16][k4 + half];
      a1.y = a_lds[buf][am + 16][k4 + half + 1];
      b0.x = b_lds[buf][bn][k4 + half];
      b0.y = b_lds[buf][bn][k4 + half + 1];
      b1.x = b_lds[buf][bn + 16][k4 + half];
      b1.y = b_lds[buf][bn + 16][k4 + half + 1];
      c00 = __builtin_amdgcn_wmma_f32_16x16x4_f32(false, a0, false, b0, (short)0, c00, false, false);
      c01 = __builtin_amdgcn_wmma_f32_16x16x4_f32(false, a0, false, b1, (short)0, c01, false, false);
      c10 = __builtin_amdgcn_wmma_f32_16x16x4_f32(false, a1, false, b0, (short)0, c10, false, false);
      c11 = __builtin_amdgcn_wmma_f32_16x16x4_f32(false, a1, false, b1, (short)0, c11, false, false);
    }
    if (more) {
      stage(buf ^ 1);                    // write the idle buffer
      __syncthreads();                   // one barrier per K chunk
      buf ^= 1;
    }
  }

  // ---- epilogue: VGPR r -> row r (lanes 0-15) / r+8 (lanes 16-31) ----
  const int rb = m0 + wm * 32 + (l >> 4) * 8;
  const int cb = n0 + wn * 32 + (l & 15);
#pragma unroll
  for (int i = 0; i < 2; ++i) {
#pragma unroll
    for (int j = 0; j < 2; ++j) {
      const v8f& cc = i ? (j ? c11 : c10) : (j ? c01 : c00);
      const int col = cb + 16 * j;
#pragma unroll
      for (int r = 0; r < 8; ++r) {
        const int row = rb + 16 * i + r;
        float v = cc[r] * scale;
        if (bias)     v += bias[col];
        if (residual) v += residual[(size_t)row * ldc + col];
        if (relu)     v = fmaxf(v, 0.0f);
        C[(size_t)row * ldc + col] = v;
      }
    }
  }
}

// ---------------------------------------------------------------------------
// Row norm (torch-style, unbiased std): alpha*(x-mean)/(std+eps)+beta, d=512
// ---------------------------------------------------------------------------
__global__ __launch_bounds__(256) void norm512_kernel(
    const float* __restrict__ x, const float* __restrict__ alpha,
    const float* __restrict__ beta, float* __restrict__ out)
{
  __shared__ float s1[256], s2[256];
  const int t = threadIdx.x;
  const size_t base = (size_t)blockIdx.x * 512;
  const float a = x[base + t];
  const float b = x[base + t + 256];
  s1[t] = a + b;
  s2[t] = a * a + b * b;
  __syncthreads();
  for (int off = 128; off > 0; off >>= 1) {
    if (t < off) { s1[t] += s1[t + off]; s2[t] += s2[t + off]; }
    __syncthreads();
  }
  const float mean = s1[0] * (1.0f / 512.0f);
  const float var  = (s2[0] - 512.0f * mean * mean) * (1.0f / 511.0f);
  const float den  = sqrtf(fmaxf(var, 0.0f)) + EPSF;
  out[base + t]       = alpha[t]       * (a - mean) / den + beta[t];
  out[base + t + 256] = alpha[t + 256] * (b - mean) / den + beta[t + 256];
}

// ---------------------------------------------------------------------------
// Column softmax over token axis (N rows), in place. Block = 32 columns.
// ---------------------------------------------------------------------------
__global__ __launch_bounds__(256) void col_softmax_kernel(float* __restrict__ k,
                                                          int nrows, int d)
{
  __shared__ float sm[8][32], ss[8][32], fm[32], fs[32];
  const int t    = threadIdx.x;
  const int lane = t & 31;
  const int rg   = t >> 5;
  const int col  = blockIdx.x * 32 + lane;

  float m = -__builtin_inff(), s = 0.0f;
  for (int r = rg; r < nrows; r += 8) {
    const float v = k[(size_t)r * d + col];
    const float nm = fmaxf(m, v);
    s = s * __expf(m - nm) + __expf(v - nm);
    m = nm;
  }
  sm[rg][lane] = m; ss[rg][lane] = s;
  __syncthreads();
  if (t < 32) {
    float M = sm[0][t], S = ss[0][t];
    for (int c = 1; c < 8; ++c) {
      const float m2 = sm[c][t], s2 = ss[c][t];
      const float nm = fmaxf(M, m2);
      S = S * __expf(M - nm) + s2 * __expf(m2 - nm);
      M = nm;
    }
    fm[t] = M; fs[t] = 1.0f / S;
  }
  __syncthreads();
  const float M = fm[lane], inv = fs[lane];
  for (int r = rg; r < nrows; r += 8) {
    const size_t idx = (size_t)r * d + col;
    k[idx] = __expf(k[idx] - M) * inv;
  }
}

// ---------------------------------------------------------------------------
// Softmax over 64 contiguous elements (query head channels), in place.
// ---------------------------------------------------------------------------
__global__ __launch_bounds__(256) void seg_softmax64_kernel(float* __restrict__ q)
{
  const int w = threadIdx.x >> 5, l = threadIdx.x & 31;
  float* p = q + ((size_t)blockIdx.x * 8 + w) * 64;
  const float a = p[l], b = p[l + 32];
  float m = fmaxf(a, b);
  for (int i = 16; i > 0; i >>= 1) m = fmaxf(m, __shfl_xor(m, i, 32));
  float s = __expf(a - m) + __expf(b - m);
  for (int i = 16; i > 0; i >>= 1) s += __shfl_xor(s, i, 32);
  const float inv = 1.0f / s;
  p[l]      = __expf(a - m) * inv;
  p[l + 32] = __expf(b - m) * inv;
}

// ---------------------------------------------------------------------------
// context[h] = key_sm[:,h,:]^T @ values[:,h,:]  (64x64 per head), split-K.
// ---------------------------------------------------------------------------
__global__ __launch_bounds__(256) void context_partial_kernel(
    const float* __restrict__ ksm, const float* __restrict__ val,
    float* __restrict__ part, int rows_per_chunk)
{
  __shared__ float sk[64], sv[64];
  const int head = blockIdx.y, chunk = blockIdx.x, t = threadIdx.x;
  const int k0 = (t >> 4) * 4, v0 = (t & 15) * 4;
  float acc[4][4] = {};
  const int nbeg = chunk * rows_per_chunk;
  for (int n = nbeg; n < nbeg + rows_per_chunk; ++n) {
    if (t < 64)       sk[t]      = ksm[(size_t)n * 512 + head * 64 + t];
    else if (t < 128) sv[t - 64] = val[(size_t)n * 512 + head * 64 + (t - 64)];
    __syncthreads();
#pragma unroll
    for (int i = 0; i < 4; ++i)
#pragma unroll
      for (int j = 0; j < 4; ++j)
        acc[i][j] += sk[k0 + i] * sv[v0 + j];
    __syncthreads();
  }
  float* dst = part + ((size_t)head * gridDim.x + chunk) * 4096;
#pragma unroll
  for (int i = 0; i < 4; ++i)
#pragma unroll
    for (int j = 0; j < 4; ++j)
      dst[(k0 + i) * 64 + v0 + j] = acc[i][j];
}

__global__ __launch_bounds__(256) void context_reduce_kernel(
    const float* __restrict__ part, float* __restrict__ ctx, int nchunks)
{
  const int idx  = blockIdx.x * 256 + threadIdx.x;   // 8*4096 total
  const int head = idx >> 12, e = idx & 4095;
  float s = 0.0f;
  for (int c = 0; c < nchunks; ++c)
    s += part[((size_t)head * nchunks + c) * 4096 + e];
  ctx[idx] = s;
}

// ---------------------------------------------------------------------------
extern "C" void kernel_launch(void* const* d_in, const int* in_sizes, int n_in,
                              void* d_out, int out_size, void* d_ws, size_t ws_size,
                              hipStream_t stream)
{
  (void)in_sizes; (void)n_in; (void)out_size; (void)ws_size;

  const float* x      = (const float*)d_in[0];
  const float* Wk     = (const float*)d_in[1];
  const float* bk     = (const float*)d_in[2];
  const float* Wq     = (const float*)d_in[3];
  const float* bq     = (const float*)d_in[4];
  const float* Wv     = (const float*)d_in[5];
  const float* bv     = (const float*)d_in[6];
  const float* Wr     = (const float*)d_in[7];
  const float* br     = (const float*)d_in[8];
  const float* alpha1 = (const float*)d_in[9];
  const float* beta1  = (const float*)d_in[10];
  const float* alpha2 = (const float*)d_in[11];
  const float* beta2  = (const float*)d_in[12];
  const float* W1     = (const float*)d_in[13];
  const float* b1     = (const float*)d_in[14];
  const float* W2     = (const float*)d_in[15];
  const float* b2     = (const float*)d_in[16];

  const int N = 8192, D = 512, F = 256;
  const size_t ND = (size_t)N * D;

  float* out_x      = (float*)d_out;          // [8192 x 512]
  float* out_scores = out_x + ND;             // [8192 x 8192]

  float* ws   = (float*)d_ws;
  float* x2   = ws;                  // [N*D]   (later reused as 'attended')
  float* keys = x2 + ND;             // [N*D]   -> key_sm in place
  float* qrs  = keys + ND;           // [N*D]   -> q_sm in place
  float* vals = qrs + ND;            // [N*D]   (later reused as x2b)
  float* part = vals + ND;           // [8*64*4096]
  float* ctx  = part + (size_t)8 * 64 * 4096; // [8*4096]
  float* h1   = ctx + 8 * 4096;      // [N*F]

  const dim3 blk(256);
  const dim3 gD(D / BN, N / BM);     // (8,64)
  const dim3 gF(F / BN, N / BM);     // (4,64)
  const dim3 gS(N / BN, N / BM);     // (128,64)

  // 1) x2 = norm1(x)
  norm512_kernel<<<N, blk, 0, stream>>>(x, alpha1, beta1, x2);

  // 2) projections
  gemm_wmma_f32<<<gD, blk, 0, stream>>>(x2, Wk, bk, nullptr, keys,
                                        N, D, D, D, D, D, 1.0f, 0);
  gemm_wmma_f32<<<gD, blk, 0, stream>>>(x2, Wq, bq, nullptr, qrs,
                                        N, D, D, D, D, D, 1.0f, 0);
  gemm_wmma_f32<<<gD, blk, 0, stream>>>(x2, Wv, bv, nullptr, vals,
                                        N, D, D, D, D, D, 1.0f, 0);

  // 3) softmaxes (in place)
  col_softmax_kernel<<<D / 32, blk, 0, stream>>>(keys, N, D);
  seg_softmax64_kernel<<<N, blk, 0, stream>>>(qrs);

  // 4) scores = (q_sm @ key_sm^T) / 8  -> NT GEMM, 256 MB output
  gemm_wmma_f32<<<gS, blk, 0, stream>>>(qrs, keys, nullptr, nullptr, out_scores,
                                        N, N, D, D, D, N, 0.125f, 1);

  // 5) context per head (deterministic split-K)
  context_partial_kernel<<<dim3(64, 8), blk, 0, stream>>>(keys, vals, part, N / 64);
  context_reduce_kernel<<<(8 * 4096) / 256, blk, 0, stream>>>(part, ctx, 64);

  // 6) attended = q_sm @ blockdiag(context)  (K=64 per head), reuse x2
  gemm_wmma_f32<<<gD, blk, 0, stream>>>(qrs, ctx, nullptr, nullptr, x2,
                                        N, D, 64, D, 64, D, 1.0f, 4);

  // 7) out_x = x + attended @ Wr + br
  gemm_wmma_f32<<<gD, blk, 0, stream>>>(x2, Wr, br, x, out_x,
                                        N, D, D, D, D, D, 1.0f, 0);

  // 8) x2b = norm2(out_x)  (reuse vals)
  norm512_kernel<<<N, blk, 0, stream>>>(out_x, alpha2, beta2, vals);

  // 9) h1 = relu(x2b @ W1 + b1)
  gemm_wmma_f32<<<gF, blk, 0, stream>>>(vals, W1, b1, nullptr, h1,
                                        N, F, D, D, F, F, 1.0f, 2);

  // 10) out_x += h1 @ W2 + b2
  gemm_wmma_f32<<<gD, blk, 0, stream>>>(h1, W2, b2, out_x, out_x,
                                        N, D, F, F, D, D, 1.0f, 0);
}